// NeuralAdditiveModel_25701084299571
// MI455X (gfx1250) — compile-verified
//
#include <hip/hip_runtime.h>

// ---------------------------------------------------------------------------
// Neural Additive Model forward for MI455X (gfx1250, wave32).
//   out[b] = bias + sum_f [ b2[f] + sum_h relu(x[b,f]*W1[f,h]+b1[f,h])*W2[f,h] ]
//
// VALU-bound problem (memory floor ~2.9us allows only ~4 lane-ops/element;
// the math needs ~34). So: minimize per-element ops (relu folded into
// |fma| via relu(t)*W2 = t*W2/2 + |t|*W2/2), keep 4 independent FMA chains
// for VOPD dual-issue, offload the feature reduction to the matrix unit
// (V_WMMA_F32_16X16X4_F32), and move all x staging onto the Tensor Data
// Mover so the VALU pipe sees almost nothing but FMAs.
// ---------------------------------------------------------------------------

typedef __attribute__((ext_vector_type(2))) float v2f;
typedef __attribute__((ext_vector_type(8))) float v8f;
typedef __attribute__((ext_vector_type(4))) unsigned int u32x4;
typedef __attribute__((ext_vector_type(8))) int i32x8;
typedef __attribute__((ext_vector_type(4))) int i32x4;

#define HIDDEN 16
#define NFEAT 1024
#define BATCH 16384

#define FG 32            // features per workgroup = 8 waves * 4 (WMMA K)
#define BT 128           // batch rows per TDM tile
#define TILES_PER_WG 8   // tiles per workgroup (1024 rows each WG)
#define XSTRIDE 33       // LDS row stride in floats (TDM pad: 32dw + 1dw)

#if __has_builtin(__builtin_amdgcn_tensor_load_to_lds) && __has_builtin(__builtin_amdgcn_s_wait_tensorcnt)
#define HAVE_TDM 1
#else
#define HAVE_TDM 0
#endif

__global__ void nam_init(float* __restrict__ out, const float* __restrict__ bias) {
  int i = blockIdx.x * blockDim.x + threadIdx.x;
  if (i < BATCH) out[i] = bias[0];
}

#if HAVE_TDM
// LDS-aperture flat pointers carry the LDS byte offset in addr[31:0] (ISA 10.2).
__device__ __forceinline__ unsigned lds_addr_of(const void* p) {
  return (unsigned)(unsigned long long)p;
}

// D# for a [BT rows x 32 floats] tile of x (row stride NFEAT floats), with
// hardware padding of 1 DWORD after every 32 DWORDs -> LDS stride 33 floats.
__device__ __forceinline__ void tdm_load_tile(const float* gsrc, unsigned lds_off) {
  unsigned long long ga = (unsigned long long)(const void*)gsrc;
  u32x4 g0;
  g0[0] = 1u;                                                // count=1, user desc
  g0[1] = lds_off;                                           // lds_addr (bytes)
  g0[2] = (unsigned)ga;                                      // global_addr[31:0]
  g0[3] = (unsigned)((ga >> 32) & 0x1FFFFFFu) | (2u << 30);  // addr[56:32] | type=2
  i32x8 g1;
  g1[0] = (2 << 16)        // data_size = 4B
        | (1 << 20)        // pad_enable
        | (4 << 22);       // pad_interval: 32 DWORDs (pad_amount=0 -> 1 DWORD)
  g1[1] = (int)(32u << 16);            // tensor_dim0 = 32 (lo16 in [31:16])
  g1[2] = (int)((unsigned)BT << 16);   // dim0 hi = 0 | tensor_dim1 = BT
  g1[3] = (int)(32u << 16);            // dim1 hi = 0 | tile_dim0 = 32
  g1[4] = BT;                          // tile_dim1 = BT, tile_dim2 = 0
  g1[5] = NFEAT;                       // tensor_dim0_stride = 1024 (lo32)
  g1[6] = 0;                           // stride hi, dim1_stride lo
  g1[7] = 0;
  i32x4 z4 = {0, 0, 0, 0};             // groups 2/3: dims 2+ unused (tile_dim2=0)
  i32x8 z8 = {0, 0, 0, 0, 0, 0, 0, 0};
  // clang-23 / therock 6-arg form: (g0, g1, g2, g3, extra, cpol)
  __builtin_amdgcn_tensor_load_to_lds(g0, g1, z4, z4, z8, 0);
}
#endif

__global__ __launch_bounds__(256)
void nam_main(const float* __restrict__ x,
              const float* __restrict__ W1, const float* __restrict__ b1,
              const float* __restrict__ W2, const float* __restrict__ b2,
              float* __restrict__ out) {
  __shared__ float lx[2][BT * XSTRIDE];   // double-buffered x tiles
  __shared__ float lout[BT];              // per-row partial sums

  const int tid  = threadIdx.x;
  const int wave = tid >> 5;
  const int lane = tid & 31;
  const int half = lane >> 4;          // WMMA half-wave
  const int m    = lane & 15;          // WMMA row (M)
  const int fgbase = blockIdx.x * FG;
  const int f0 = fgbase + wave * 4 + 2 * half;   // lane reg r -> WMMA K=2*half+r

  // ---- per-lane weight transform (held in VGPRs for the whole WG) ----
  float w12[2][HIDDEN], b12[2][HIDDEN], sgn[2][HIDDEN], SW[2], SB[2];
#pragma unroll
  for (int r = 0; r < 2; ++r) {
    const int f = f0 + r;
    float sw = 0.0f, sb = 0.0f;
#pragma unroll
    for (int h = 0; h < HIDDEN; ++h) {
      float w1v = W1[f * HIDDEN + h];
      float b1v = b1[f * HIDDEN + h];
      float w2v = W2[f * HIDDEN + h];
      float w2h = 0.5f * w2v;
      w12[r][h] = w1v * w2h;
      b12[r][h] = b1v * w2h;
      sgn[r][h] = (w2v >= 0.0f) ? 1.0f : -1.0f;
      sw += w12[r][h];
      sb += b12[r][h];
    }
    SW[r] = sw;
    SB[r] = sb + b2[f];
  }

  if (tid < BT) lout[tid] = 0.0f;

  // B matrix: ones in column N=0 (lanes 0 and 16 per ISA B layout), else 0.
  v2f Bm;
  Bm.x = (m == 0) ? 1.0f : 0.0f;
  Bm.y = Bm.x;

  const int cbase = wave * 4 + 2 * half;
  const int tile0 = blockIdx.y * TILES_PER_WG;

#if HAVE_TDM
  // Prefetch pipeline: keep up to two tiles in flight on the TDM.
  if (wave == 0) {
    tdm_load_tile(x + (size_t)(tile0 + 0) * BT * NFEAT + fgbase,
                  lds_addr_of(&lx[0][0]));
    tdm_load_tile(x + (size_t)(tile0 + 1) * BT * NFEAT + fgbase,
                  lds_addr_of(&lx[1][0]));
  }
#endif

  for (int bt = 0; bt < TILES_PER_WG; ++bt) {
    const int rowbase = (tile0 + bt) * BT;
    const int b = bt & 1;

#if HAVE_TDM
    if (wave == 0) {
      if (bt + 1 < TILES_PER_WG) __builtin_amdgcn_s_wait_tensorcnt(1);
      else                       __builtin_amdgcn_s_wait_tensorcnt(0);
    }
#else
    // Fallback: synchronous cooperative staging (coalesced 128B wave loads).
#pragma unroll
    for (int i = 0; i < BT / 8; ++i) {
      int rr = wave + 8 * i;
      lx[b][rr * XSTRIDE + lane] = x[(size_t)(rowbase + rr) * NFEAT + fgbase + lane];
    }
#endif
    __syncthreads();   // tile bt visible to all waves; lout zeroed

    const float* lb = &lx[b][0];
    for (int rt = 0; rt < BT / 16; ++rt) {
      const int row = rt * 16 + m;
      float xv0 = lb[row * XSTRIDE + cbase + 0];
      float xv1 = lb[row * XSTRIDE + cbase + 1];

      // 4 independent FMA chains (even/odd h per feature) for dual-issue.
      float a0a = fmaf(xv0, SW[0], SB[0]);
      float a0b = 0.0f;
      float a1a = fmaf(xv1, SW[1], SB[1]);
      float a1b = 0.0f;
#pragma unroll
      for (int h = 0; h < HIDDEN; h += 2) {
        float u0 = fmaf(xv0, w12[0][h], b12[0][h]);
        a0a = fmaf(sgn[0][h], fabsf(u0), a0a);       // v_fma_f32 |src| mod
        float u1 = fmaf(xv0, w12[0][h + 1], b12[0][h + 1]);
        a0b = fmaf(sgn[0][h + 1], fabsf(u1), a0b);
        float u2 = fmaf(xv1, w12[1][h], b12[1][h]);
        a1a = fmaf(sgn[1][h], fabsf(u2), a1a);
        float u3 = fmaf(xv1, w12[1][h + 1], b12[1][h + 1]);
        a1b = fmaf(sgn[1][h + 1], fabsf(u3), a1b);
      }

      // WMMA reduction over the wave's 4 features (K dim): D[m,0]=sum_k A[m,k]
      v2f A;
      A.x = a0a + a0b;   // K = 2*half + 0
      A.y = a1a + a1b;   // K = 2*half + 1
      v8f C = {};
      C = __builtin_amdgcn_wmma_f32_16x16x4_f32(
          false, A, false, Bm, (short)0, C, false, false);

      // D[m,0]: lane 0 holds rows 0..7 (VGPR j), lane 16 holds rows 8..15.
      if (m == 0) {
        int rb = rt * 16 + half * 8;
#pragma unroll
        for (int j = 0; j < 8; ++j)
          atomicAdd(&lout[rb + j], C[j]);            // ds_add_f32
      }
    }
    __syncthreads();   // all waves done reading lx[b] and adding to lout

    if (tid < BT) {
      unsafeAtomicAdd(&out[rowbase + tid], lout[tid]);  // 32 adds/out elem total
      lout[tid] = 0.0f;
    }

#if HAVE_TDM
    if (wave == 0 && bt + 2 < TILES_PER_WG)
      tdm_load_tile(x + (size_t)(tile0 + bt + 2) * BT * NFEAT + fgbase,
                    lds_addr_of(&lx[b][0]));
#endif
  }
}

extern "C" void kernel_launch(void* const* d_in, const int* in_sizes, int n_in,
                              void* d_out, int out_size, void* d_ws, size_t ws_size,
                              hipStream_t stream) {
  (void)in_sizes; (void)n_in; (void)out_size; (void)d_ws; (void)ws_size;
  const float* x    = (const float*)d_in[0];
  const float* W1   = (const float*)d_in[1];
  const float* b1   = (const float*)d_in[2];
  const float* W2   = (const float*)d_in[3];
  const float* b2   = (const float*)d_in[4];
  const float* bias = (const float*)d_in[5];
  float* out = (float*)d_out;

  nam_init<<<(BATCH + 255) / 256, 256, 0, stream>>>(out, bias);

  dim3 grid(NFEAT / FG, BATCH / (BT * TILES_PER_WG));  // 32 x 16 = 512 WGs
  nam_main<<<grid, 256, 0, stream>>>(x, W1, b1, W2, b2, out);
}